// SNNController_17609365914544
// MI455X (gfx1250) — compile-verified
//
#include <hip/hip_runtime.h>

// SNN forward: T sequential LIF steps, batch-parallel, fp32 WMMA on gfx1250.
// v2: bank-conflict-free padded LDS, double-buffered X fragments, 4-way
//     split-K ILP in layer 2, non-temporal output stores.

typedef __attribute__((ext_vector_type(2))) float v2f;
typedef __attribute__((ext_vector_type(4))) float v4f;
typedef __attribute__((ext_vector_type(8))) float v8f;

#define T_STEPS 1024
#define BATCH   512
#define IN_DIM  64
#define HID_DIM 256
#define OUT_DIM 32
#define BETA    0.92f
#define TH      1.0f

// Padded LDS row strides (in floats). Stride % 64 == 4 makes every 32-lane
// b64 fragment read touch all 64 banks exactly once (bank = 4*lane + 4k + 2*half).
#define W1_STRIDE  (IN_DIM + 4)    // 68
#define W2_STRIDE  (HID_DIM + 4)   // 260
#define SPK_STRIDE (HID_DIM + 4)   // 260

__launch_bounds__(128, 1)
__global__ void snn_lif_kernel(const float* __restrict__ x,   // [T,B,IN]
                               const float* __restrict__ W1,  // [HID,IN]
                               const float* __restrict__ b1,  // [HID]
                               const float* __restrict__ W2,  // [OUT,HID]
                               const float* __restrict__ b2,  // [OUT]
                               float* __restrict__ out)       // [2,T,B,OUT]
{
    __shared__ float lds_w1[HID_DIM * W1_STRIDE];   // ~68 KB
    __shared__ float lds_w2[OUT_DIM * W2_STRIDE];   // ~32.5 KB
    __shared__ float lds_spk1[16 * SPK_STRIDE];     // ~16.3 KB

    const int tid   = threadIdx.x;     // 0..127
    const int wave  = tid >> 5;        // 0..3
    const int lane  = tid & 31;
    const int lhalf = lane >> 4;       // K-half selector for A/B fragments
    const int lmod  = lane & 15;       // M (A) / N (B,C) index within tile
    const int b0    = blockIdx.x * 16; // batch row base of this workgroup

    // ---- preload weights into padded LDS (vectorized, all 128 threads) ----
    for (int i = tid; i < HID_DIM * (IN_DIM / 4); i += 128) {      // 4096 v4f
        const int row = i >> 4, col = (i & 15) * 4;
        *(v4f*)(lds_w1 + row * W1_STRIDE + col) =
            *(const v4f*)(W1 + row * IN_DIM + col);
    }
    for (int i = tid; i < OUT_DIM * (HID_DIM / 4); i += 128) {     // 2048 v4f
        const int row = i >> 6, col = (i & 63) * 4;
        *(v4f*)(lds_w2 + row * W2_STRIDE + col) =
            *(const v4f*)(W2 + row * HID_DIM + col);
    }

    // per-lane biases: C/D layout has N = lane%16 constant across all 8 rows
    float bias1[4];
#pragma unroll
    for (int nt = 0; nt < 4; ++nt)
        bias1[nt] = b1[(wave * 4 + nt) * 16 + lmod];
    const float bias2 = b2[(wave & 1) * 16 + lmod];

    __syncthreads();

    // persistent membrane state (lives in VGPRs across all T steps)
    v8f mem1[4];
#pragma unroll
    for (int nt = 0; nt < 4; ++nt) mem1[nt] = (v8f){};
    v8f mem2 = (v8f){};

    // A-fragment base: X[b0+lmod][2*lhalf + ...]
    const float* xrow = x + (size_t)(b0 + lmod) * IN_DIM + 2 * lhalf;

    // double-buffered A fragments (16 K-steps of 4 each)
    v2f afA[16], afB[16];
#pragma unroll
    for (int kk = 0; kk < 16; ++kk)
        afA[kk] = *(const v2f*)(xrow + 4 * kk);    // prologue: t = 0

    auto step = [&](int t, v2f (&af)[16], v2f (&afn)[16]) {
        // ---- prefetch next timestep's A fragments (overlaps with WMMAs) ----
        const int tn = (t + 1 < T_STEPS) ? (t + 1) : t;
        const float* xnext = xrow + (size_t)tn * (BATCH * IN_DIM);
#pragma unroll
        for (int kk = 0; kk < 16; ++kk)
            afn[kk] = *(const v2f*)(xnext + 4 * kk);

        // ---- layer 1: each wave computes 4 hid N-tiles (4 indep. chains) ----
#pragma unroll
        for (int nt = 0; nt < 4; ++nt) {
            const int n0 = (wave * 4 + nt) * 16;
            v8f acc = (v8f){};
            const float* wrow = lds_w1 + (n0 + lmod) * W1_STRIDE + 2 * lhalf;
#pragma unroll
            for (int kk = 0; kk < 16; ++kk) {
                v2f bfrag = *(const v2f*)(wrow + 4 * kk);
                acc = __builtin_amdgcn_wmma_f32_16x16x4_f32(
                        false, af[kk], false, bfrag, (short)0, acc,
                        false, false);
            }
            // LIF update (reset from previous membrane, subtract mechanism)
            v8f spk;
#pragma unroll
            for (int r = 0; r < 8; ++r) {
                float cur   = acc[r] + bias1[nt];
                float m     = mem1[nt][r];
                float reset = (m > TH) ? TH : 0.0f;
                m           = BETA * m + cur - reset;
                mem1[nt][r] = m;
                spk[r]      = (m > TH) ? 1.0f : 0.0f;
            }
            // scatter spikes to LDS: row = r + 8*lhalf, col = n0 + lmod
#pragma unroll
            for (int r = 0; r < 8; ++r)
                lds_spk1[(r + 8 * lhalf) * SPK_STRIDE + n0 + lmod] = spk[r];
        }
        __syncthreads();

        // ---- layer 2: waves 0,1 one 16x16 tile each; 4 split-K chains ----
        if (wave < 2) {
            const int n0 = wave * 16;
            v8f acc0 = (v8f){}, acc1 = (v8f){}, acc2 = (v8f){}, acc3 = (v8f){};
            const float* arow = lds_spk1 + lmod * SPK_STRIDE + 2 * lhalf;
            const float* wrow = lds_w2 + (n0 + lmod) * W2_STRIDE + 2 * lhalf;
#pragma unroll 4
            for (int j = 0; j < 16; ++j) {
                v2f a0 = *(const v2f*)(arow + 4 * (j));
                v2f b0f = *(const v2f*)(wrow + 4 * (j));
                v2f a1 = *(const v2f*)(arow + 4 * (16 + j));
                v2f b1f = *(const v2f*)(wrow + 4 * (16 + j));
                v2f a2 = *(const v2f*)(arow + 4 * (32 + j));
                v2f b2f = *(const v2f*)(wrow + 4 * (32 + j));
                v2f a3 = *(const v2f*)(arow + 4 * (48 + j));
                v2f b3f = *(const v2f*)(wrow + 4 * (48 + j));
                acc0 = __builtin_amdgcn_wmma_f32_16x16x4_f32(
                        false, a0, false, b0f, (short)0, acc0, false, false);
                acc1 = __builtin_amdgcn_wmma_f32_16x16x4_f32(
                        false, a1, false, b1f, (short)0, acc1, false, false);
                acc2 = __builtin_amdgcn_wmma_f32_16x16x4_f32(
                        false, a2, false, b2f, (short)0, acc2, false, false);
                acc3 = __builtin_amdgcn_wmma_f32_16x16x4_f32(
                        false, a3, false, b3f, (short)0, acc3, false, false);
            }
            const v8f acc = (acc0 + acc1) + (acc2 + acc3);

            // LIF + non-temporal global store of spk2 / mem2 (streamed once)
            const size_t obase   = (size_t)t * BATCH * OUT_DIM;
            const size_t mem_off = (size_t)T_STEPS * BATCH * OUT_DIM;
#pragma unroll
            for (int r = 0; r < 8; ++r) {
                float cur   = acc[r] + bias2;
                float m     = mem2[r];
                float reset = (m > TH) ? TH : 0.0f;
                m           = BETA * m + cur - reset;
                mem2[r]     = m;
                float s     = (m > TH) ? 1.0f : 0.0f;
                const int row = b0 + r + 8 * lhalf;
                const size_t idx = obase + (size_t)row * OUT_DIM + n0 + lmod;
                __builtin_nontemporal_store(s, &out[idx]);           // spk2_rec
                __builtin_nontemporal_store(m, &out[mem_off + idx]); // mem2_rec
            }
        }
        __syncthreads();  // protect lds_spk1 before next timestep's writes
    };

    for (int t = 0; t < T_STEPS; t += 2) {
        step(t,     afA, afB);   // compute with A, prefetch into B
        step(t + 1, afB, afA);   // compute with B, prefetch into A
    }
}

extern "C" void kernel_launch(void* const* d_in, const int* in_sizes, int n_in,
                              void* d_out, int out_size, void* d_ws, size_t ws_size,
                              hipStream_t stream) {
    (void)in_sizes; (void)n_in; (void)out_size; (void)d_ws; (void)ws_size;
    const float* x  = (const float*)d_in[0];
    const float* W1 = (const float*)d_in[1];
    const float* b1 = (const float*)d_in[2];
    const float* W2 = (const float*)d_in[3];
    const float* b2 = (const float*)d_in[4];
    float* out = (float*)d_out;

    dim3 grid(BATCH / 16);   // 32 workgroups, one 16-row batch tile each
    dim3 block(128);         // 4 waves of 32
    hipLaunchKernelGGL(snn_lif_kernel, grid, block, 0, stream,
                       x, W1, b1, W2, b2, out);
}